// ParticleI2cCell_9818295239340
// MI455X (gfx1250) — compile-verified
//
#include <hip/hip_runtime.h>
#include <math.h>

// ---------------------------------------------------------------------------
// ParticleI2cCell smoothing update, fused for MI455X (gfx1250, wave32).
//
// Math (exact rewrite of reference; -0.5|s_m|^2 - log_const cancels):
//   mu_p  = A x_p + B u_p
//   c_p   = log_weights[p] - 0.5|mu_p|^2
//   t[m,p] = s_m . mu_p
//   L[m]  = logsumexp_p ( c_p + t[m,p] )
//   out[p]= logsumexp_m ( t[m,p] + c_p + (weights[m] - L[m]) )
//
// Both passes are the SAME tiled 8192x8192 (K=20) f32 WMMA GEMM with a fused
// flash-style online logsumexp over the WMMA M axis.  Biases ride in padded
// K slots.  MUX rows are pre-scaled by log2(e) so WMMA outputs are already in
// the base-2 domain: the hot loop uses raw v_exp_f32 (no per-element mul).
// ---------------------------------------------------------------------------

typedef float v2f __attribute__((ext_vector_type(2)));
typedef float v8f __attribute__((ext_vector_type(8)));

#define NP 8192
#define MM 8192
#define DXD 16
#define DUD 8
#define KP 20            // padded K (16 data + c + 1 + 2 zero)
#define NCT 4            // 16-column tiles per block  -> 64 cols/block
#define NWAVES 8         // 256 threads/block

#define LOG2E 1.4426950408889634f
#define LN2   0.6931471805599453f

// MUX[p] = log2e * [mu_p(16), c_p, 1, 0, 0] ; SV[m] = [s_m(16), 1, 0, 0, 0].
// dot(MUX_p, SV_m) = log2e * (t[m,p] + c_p + bias_m)   (base-2 domain)
__global__ __launch_bounds__(128) void prep_kernel(
    const float* __restrict__ particles, const float* __restrict__ log_weights,
    const float* __restrict__ samples_p, const float* __restrict__ A,
    const float* __restrict__ B, float* __restrict__ MUX, float* __restrict__ SV)
{
  int i = blockIdx.x * blockDim.x + threadIdx.x;
  if (i < NP) {
    const float* row = particles + i * (DXD + DUD);
    float x[DXD], u[DUD];
    #pragma unroll
    for (int k = 0; k < DXD; k++) x[k] = row[k];
    #pragma unroll
    for (int k = 0; k < DUD; k++) u[k] = row[DXD + k];
    float* o = MUX + i * KP;
    float ss = 0.f;
    #pragma unroll
    for (int j = 0; j < DXD; j++) {
      float m = 0.f;
      #pragma unroll
      for (int k = 0; k < DXD; k++) m = fmaf(A[j * DXD + k], x[k], m);
      #pragma unroll
      for (int k = 0; k < DUD; k++) m = fmaf(B[j * DUD + k], u[k], m);
      o[j] = LOG2E * m;
      ss = fmaf(m, m, ss);
    }
    o[16] = LOG2E * (log_weights[i] - 0.5f * ss);  // log2e * c_p
    o[17] = LOG2E;                                 // picks up bias_m (natural)
    o[18] = 0.f; o[19] = 0.f;
  }
  if (i < MM) {
    float* o = SV + i * KP;
    const float* s = samples_p + i * DXD;
    #pragma unroll
    for (int j = 0; j < DXD; j++) o[j] = s[j];
    o[16] = 1.0f;   // picks up log2e*c_p from MUX[16]
    o[17] = 0.0f;   // bias_m (natural units), filled after pass 1
    o[18] = 0.f; o[19] = 0.f;
  }
}

__global__ __launch_bounds__(128) void bias_kernel(
    const float* __restrict__ weights, const float* __restrict__ L,
    float* __restrict__ SV)
{
  int m = blockIdx.x * blockDim.x + threadIdx.x;
  if (m < MM) SV[m * KP + 17] = weights[m] - L[m];
}

// out[col] = ln2 * log2-sum-exp2 over all rows of dot(Amat[row], Bmat[col]).
__global__ __launch_bounds__(NWAVES * 32) void lse_gemm_kernel(
    const float* __restrict__ Amat,   // [nrows][KP] (reduced axis)
    const float* __restrict__ Bmat,   // [ncols][KP] (output axis)
    float* __restrict__ out,          // [ncols], natural-log domain
    int nrow_tiles)
{
  const int tid  = threadIdx.x;
  const int lane = tid & 31;
  const int wave = tid >> 5;
  const int half = (lane >> 4) & 1;
  const int l16  = lane & 15;
  const int koff = half * 2;          // lanes 0-15 hold K pair {0,1}, 16-31 {2,3}

  const int col_base = blockIdx.x * (NCT * 16);

  // Preload B operands (one 16-col tile each) for all NCT tiles: 5 k-chunks.
  v2f breg[NCT][5];
  #pragma unroll
  for (int ct = 0; ct < NCT; ct++) {
    const float* bp = Bmat + (size_t)(col_base + ct * 16 + l16) * KP + koff;
    #pragma unroll
    for (int q = 0; q < 5; q++) breg[ct][q] = *(const v2f*)(bp + 4 * q);
  }

  // Online base-2 logsumexp state per column tile.
  float rmax[NCT], rsum[NCT];
  #pragma unroll
  for (int ct = 0; ct < NCT; ct++) { rmax[ct] = -__builtin_inff(); rsum[ct] = 0.f; }

  for (int rt = wave; rt < nrow_tiles; rt += NWAVES) {
    const float* ap = Amat + (size_t)(rt * 16 + l16) * KP + koff;
    v2f areg[5];
    #pragma unroll
    for (int q = 0; q < 5; q++) areg[q] = *(const v2f*)(ap + 4 * q);

    #pragma unroll
    for (int ct = 0; ct < NCT; ct++) {
      v8f acc = {0.f, 0.f, 0.f, 0.f, 0.f, 0.f, 0.f, 0.f};
      #pragma unroll
      for (int q = 0; q < 5; q++)
        acc = __builtin_amdgcn_wmma_f32_16x16x4_f32(
            false, areg[q], false, breg[ct][q], (short)0, acc, false, false);

      // acc[j] (base-2 domain) = row (rt*16 + 8*half + j), col (col_base+ct*16+l16)
      float m01 = fmaxf(acc[0], acc[1]), m23 = fmaxf(acc[2], acc[3]);
      float m45 = fmaxf(acc[4], acc[5]), m67 = fmaxf(acc[6], acc[7]);
      float tmax = fmaxf(fmaxf(m01, m23), fmaxf(m45, m67));
      float nmax = fmaxf(rmax[ct], tmax);

      // 8 independent exp2s, then tree sum (keeps the trans pipe streaming).
      float e0 = __builtin_amdgcn_exp2f(acc[0] - nmax);
      float e1 = __builtin_amdgcn_exp2f(acc[1] - nmax);
      float e2 = __builtin_amdgcn_exp2f(acc[2] - nmax);
      float e3 = __builtin_amdgcn_exp2f(acc[3] - nmax);
      float e4 = __builtin_amdgcn_exp2f(acc[4] - nmax);
      float e5 = __builtin_amdgcn_exp2f(acc[5] - nmax);
      float e6 = __builtin_amdgcn_exp2f(acc[6] - nmax);
      float e7 = __builtin_amdgcn_exp2f(acc[7] - nmax);
      float scale = __builtin_amdgcn_exp2f(rmax[ct] - nmax);
      float s01 = e0 + e1, s23 = e2 + e3, s45 = e4 + e5, s67 = e6 + e7;
      float tsum = (s01 + s23) + (s45 + s67);
      rsum[ct] = fmaf(rsum[ct], scale, tsum);
      rmax[ct] = nmax;
    }
  }

  // Merge lane l with partner l^16 (other 8 rows of same column), then waves.
  __shared__ float smax[NWAVES][NCT][16];
  __shared__ float ssum[NWAVES][NCT][16];
  #pragma unroll
  for (int ct = 0; ct < NCT; ct++) {
    float pmax = __shfl_xor(rmax[ct], 16, 32);
    float psum = __shfl_xor(rsum[ct], 16, 32);
    float nmax = fmaxf(rmax[ct], pmax);
    float s = rsum[ct] * __builtin_amdgcn_exp2f(rmax[ct] - nmax)
            + psum    * __builtin_amdgcn_exp2f(pmax    - nmax);
    if (half == 0) { smax[wave][ct][l16] = nmax; ssum[wave][ct][l16] = s; }
  }
  __syncthreads();

  if (tid < NCT * 16) {
    int ct = tid >> 4, c = tid & 15;
    float nmax = -__builtin_inff();
    #pragma unroll
    for (int w = 0; w < NWAVES; w++) nmax = fmaxf(nmax, smax[w][ct][c]);
    float s = 0.f;
    #pragma unroll
    for (int w = 0; w < NWAVES; w++)
      s += ssum[w][ct][c] * __builtin_amdgcn_exp2f(smax[w][ct][c] - nmax);
    // convert base-2 logsumexp back to natural log
    out[col_base + ct * 16 + c] = LN2 * (nmax + __builtin_amdgcn_logf(s));
  }
}

extern "C" void kernel_launch(void* const* d_in, const int* in_sizes, int n_in,
                              void* d_out, int out_size, void* d_ws, size_t ws_size,
                              hipStream_t stream) {
  const float* particles   = (const float*)d_in[0];
  const float* log_weights = (const float*)d_in[1];
  const float* samples_p   = (const float*)d_in[2];
  const float* weights     = (const float*)d_in[3];
  const float* A           = (const float*)d_in[4];
  const float* B           = (const float*)d_in[5];
  // d_in[6] (samples_idx) is unused by the reference.

  float* MUX = (float*)d_ws;           // NP * KP
  float* SV  = MUX + (size_t)NP * KP;  // MM * KP
  float* L   = SV + (size_t)MM * KP;   // MM

  prep_kernel<<<(NP + 127) / 128, 128, 0, stream>>>(
      particles, log_weights, samples_p, A, B, MUX, SV);

  // Pass 1: reduce over particles p (rows = MUX), columns = samples m -> L[m]
  lse_gemm_kernel<<<MM / (NCT * 16), NWAVES * 32, 0, stream>>>(MUX, SV, L, NP / 16);

  bias_kernel<<<(MM + 127) / 128, 128, 0, stream>>>(weights, L, SV);

  // Pass 2: reduce over samples m (rows = SV), columns = particles p -> out[p]
  lse_gemm_kernel<<<NP / (NCT * 16), NWAVES * 32, 0, stream>>>(
      SV, MUX, (float*)d_out, MM / 16);
}